// AtomAttention_73031623901260
// MI455X (gfx1250) — compile-verified
//
#include <hip/hip_runtime.h>

typedef __attribute__((ext_vector_type(2))) float v2f;
typedef __attribute__((ext_vector_type(4))) float v4f;
typedef __attribute__((ext_vector_type(8))) float v8f;

#define D_DIM 256

// ---------------------------------------------------------------------------
// Kernel 1: v[j] = dot(solvent[j, :], w[D:2D]) via V_WMMA_F32_16X16X4_F32.
// One wave computes 16 rows. B-frag replicates the 4 weight values across all
// 16 columns, so every column of the C tile equals the row's dot product.
//
// fp32 A 16x4 layout (ISA 7.12.2): lanes 0-15 hold M=lane with K={0,1} in
// VGPR0/1; lanes 16-31 hold M=lane-16 with K={2,3}. B 4x16 mirrors: lane-half
// selects the K pair, VGPR index selects within the pair, lane&15 is N.
// ---------------------------------------------------------------------------
__global__ void gemv_wmma(const float* __restrict__ solvent,
                          const float* __restrict__ attn_w,
                          float* __restrict__ v) {
    const int lane  = threadIdx.x & 31;
    const int wave  = threadIdx.x >> 5;
    const int r0    = blockIdx.x * 128 + wave * 16;      // 8 waves * 16 rows
    const int khalf = (lane < 16) ? 0 : 2;
    const float* __restrict__ arow = solvent + (size_t)(r0 + (lane & 15)) * D_DIM;
    const float* __restrict__ w2   = attn_w + D_DIM;     // second half of weights

    v8f c = {};
#pragma unroll 4
    for (int k = 0; k < D_DIM; k += 4) {
        v2f a, b;
        a.x = arow[k + khalf];
        a.y = arow[k + khalf + 1];
        b.x = w2[k + khalf];
        b.y = w2[k + khalf + 1];
        // D = A*B + C   (fp32, exact — no precision loss before the exp)
        c = __builtin_amdgcn_wmma_f32_16x16x4_f32(
                /*neg_a=*/false, a, /*neg_b=*/false, b,
                /*c_mod=*/(short)0, c, /*reuse_a=*/false, /*reuse_b=*/false);
    }

    // C layout: VGPR r, lanes 0-15 -> (M=r, N=lane); lanes 16-31 -> (M=8+r, N=lane-16).
    // All columns are equal, so lane r reads (M=r,N=r), lane 16+r reads (M=8+r,N=r).
#pragma unroll
    for (int r = 0; r < 8; ++r) {
        if (lane == r)      v[r0 + r]     = c[r];
        if (lane == 16 + r) v[r0 + 8 + r] = c[r];
    }
}

// ---------------------------------------------------------------------------
// Kernel 2: s = softmax(v) over the full M=8192 vector, one 1024-thread block.
// (The row-constant u[i]+b cancels inside softmax, so one row serves all N.)
// ---------------------------------------------------------------------------
__global__ void softmax1d(const float* __restrict__ v, float* __restrict__ s) {
    __shared__ float red[1024];
    const int tid = threadIdx.x;

    float vals[8];
    float mx = -INFINITY;
#pragma unroll
    for (int i = 0; i < 8; ++i) {
        vals[i] = v[tid + i * 1024];
        mx = fmaxf(mx, vals[i]);
    }
    red[tid] = mx;
    __syncthreads();
    for (int off = 512; off > 0; off >>= 1) {
        if (tid < off) red[tid] = fmaxf(red[tid], red[tid + off]);
        __syncthreads();
    }
    mx = red[0];
    __syncthreads();

    float e[8];
    float sum = 0.0f;
#pragma unroll
    for (int i = 0; i < 8; ++i) {
        e[i] = expf(vals[i] - mx);
        sum += e[i];
    }
    red[tid] = sum;
    __syncthreads();
    for (int off = 512; off > 0; off >>= 1) {
        if (tid < off) red[tid] += red[tid + off];
        __syncthreads();
    }
    const float inv = 1.0f / red[0];
#pragma unroll
    for (int i = 0; i < 8; ++i) s[tid + i * 1024] = e[i] * inv;
}

// ---------------------------------------------------------------------------
// Kernel 3: broadcast s to every row of out. This is the HBM-bound part
// (256 MB of stores, ~11 us floor at 23.3 TB/s). Non-temporal 128-bit stores:
// the output exceeds the 192 MB L2 and is never re-read, while the 32 KB s
// vector stays cache-resident for the reads. Each value is held in registers
// for 32 rows; each wave stores 512 B contiguous per row iteration.
// ---------------------------------------------------------------------------
__global__ void bcast(const v4f* __restrict__ s4, v4f* __restrict__ out4,
                      int Mdiv4) {
    const int j4   = blockIdx.x * blockDim.x + threadIdx.x;
    const int row0 = blockIdx.y * 32;
    const v4f val  = s4[j4];
#pragma unroll
    for (int r = 0; r < 32; ++r) {
        __builtin_nontemporal_store(val, &out4[(size_t)(row0 + r) * Mdiv4 + j4]);
    }
}

// ---------------------------------------------------------------------------
extern "C" void kernel_launch(void* const* d_in, const int* in_sizes, int n_in,
                              void* d_out, int out_size, void* d_ws, size_t ws_size,
                              hipStream_t stream) {
    // inputs: [0] solute (unused — cancels in softmax), [1] solvent,
    //         [2] attn_w (2D), [3] attn_b (cancels in softmax)
    const float* solvent = (const float*)d_in[1];
    const float* attn_w  = (const float*)d_in[2];
    const int N = in_sizes[0] / D_DIM;   // 8192
    const int M = in_sizes[1] / D_DIM;   // 8192

    float* v = (float*)d_ws;             // [M]
    float* s = v + M;                    // [M]

    gemv_wmma<<<M / 128, 256, 0, stream>>>(solvent, attn_w, v);
    softmax1d<<<1, 1024, 0, stream>>>(v, s);

    dim3 grid((M / 4) / 256, N / 32);    // (8, 256)
    bcast<<<grid, 256, 0, stream>>>((const v4f*)s, (v4f*)d_out, M / 4);
}